// KeyRecorder_57329223467566
// MI455X (gfx1250) — compile-verified
//
#include <hip/hip_runtime.h>

typedef __attribute__((ext_vector_type(2))) float v2f;
typedef __attribute__((ext_vector_type(8))) float v8f;

#define NB 32
#define NT 4096
#define ND 512
#define NR 64
#define NLOCAL 20
#define NSAMPLED 408      // indices 0..4070 step 10
#define SAMPLED_TILES 26  // ceil(408/16)
#define TOTAL_TILES 28    // + 2 present tiles (20 rows padded to 32)
#define LN_EPS 1e-5f

static __device__ __forceinline__ v8f wmma_f32_4(v2f a, v2f b, v8f c) {
  return __builtin_amdgcn_wmma_f32_16x16x4_f32(false, a, false, b, (short)0, c,
                                               false, false);
}

// ---------------------------------------------------------------------------
// Kernel 1: cf = LN(relu(obs @ w_red + b_red)) at the 428 needed frames.
// One block per batch, 8 waves. Each wave processes TWO 16-frame tiles per
// pass so each w_red fragment fetch feeds 8 WMMAs (B-traffic halved).
// Produces past_pooled (max over sampled frames), present rows, and the
// now_reduc_max output.
// ---------------------------------------------------------------------------
__global__ __launch_bounds__(256) void k_reduce(
    const float* __restrict__ obs, const float* __restrict__ w_red,
    const float* __restrict__ b_red, const float* __restrict__ g_red,
    const float* __restrict__ beta_red,
    float* __restrict__ ws_past,      // [NB][NR]
    float* __restrict__ ws_present,   // [NB][NLOCAL][NR]
    float* __restrict__ out_now)      // [NB][NR]
{
  __shared__ float smax[8 * NR];

  const int b    = blockIdx.x;
  const int tid  = threadIdx.x;
  const int wave = tid >> 5;
  const int lane = tid & 31;
  const int half = lane >> 4;
  const int ln16 = lane & 15;

  float br[4], gr[4], betar[4];
#pragma unroll
  for (int nt = 0; nt < 4; ++nt) {
    const int r = nt * 16 + ln16;
    br[nt] = b_red[r]; gr[nt] = g_red[r]; betar[nt] = beta_red[r];
  }

  float lmax[4];
#pragma unroll
  for (int nt = 0; nt < 4; ++nt) lmax[nt] = -__builtin_inff();

  const float* obs_b = obs + (size_t)b * NT * ND;

  // frame (row) index for this lane's A-row of a given tile (clamped; pads
  // duplicate valid frames, which is max-safe and store-guarded)
  auto frame_of = [&](int tile) -> int {
    if (tile < SAMPLED_TILES) {
      int i = tile * 16 + ln16;
      if (i > NSAMPLED - 1) i = NSAMPLED - 1;
      return i * 10;
    }
    int l = (tile - SAMPLED_TILES) * 16 + ln16;
    if (l > NLOCAL - 1) l = NLOCAL - 1;
    return (NT - NLOCAL) + l;
  };

  // epilogue: bias + relu + LayerNorm(64) per row, then max-reduce or store
  auto epilogue = [&](const v8f acc[4], int tile) {
    const bool sampled = (tile < SAMPLED_TILES);
#pragma unroll
    for (int i = 0; i < 8; ++i) {
      float x0 = fmaxf(acc[0][i] + br[0], 0.f);
      float x1 = fmaxf(acc[1][i] + br[1], 0.f);
      float x2 = fmaxf(acc[2][i] + br[2], 0.f);
      float x3 = fmaxf(acc[3][i] + br[3], 0.f);
      float s  = x0 + x1 + x2 + x3;
      float ss = x0*x0 + x1*x1 + x2*x2 + x3*x3;
#pragma unroll
      for (int d = 1; d < 16; d <<= 1) {   // 16-lane segments == row halves
        s  += __shfl_xor(s,  d, 32);
        ss += __shfl_xor(ss, d, 32);
      }
      const float mean = s * (1.f / 64.f);
      const float var  = ss * (1.f / 64.f) - mean * mean;
      const float rinv = rsqrtf(var + LN_EPS);
      const float y0 = (x0 - mean) * rinv * gr[0] + betar[0];
      const float y1 = (x1 - mean) * rinv * gr[1] + betar[1];
      const float y2 = (x2 - mean) * rinv * gr[2] + betar[2];
      const float y3 = (x3 - mean) * rinv * gr[3] + betar[3];
      if (sampled) {
        lmax[0] = fmaxf(lmax[0], y0); lmax[1] = fmaxf(lmax[1], y1);
        lmax[2] = fmaxf(lmax[2], y2); lmax[3] = fmaxf(lmax[3], y3);
      } else {
        const int l = (tile - SAMPLED_TILES) * 16 + i + 8 * half;
        if (l < NLOCAL) {
          float* dst = ws_present + ((size_t)(b * NLOCAL + l)) * NR;
          dst[ 0 + ln16] = y0; dst[16 + ln16] = y1;
          dst[32 + ln16] = y2; dst[48 + ln16] = y3;
          if (l == NLOCAL - 1) {
            float* dn = out_now + (size_t)b * NR;
            dn[ 0 + ln16] = y0; dn[16 + ln16] = y1;
            dn[32 + ln16] = y2; dn[48 + ln16] = y3;
          }
        }
      }
    }
  };

  const v8f vz = {0.f,0.f,0.f,0.f,0.f,0.f,0.f,0.f};

#pragma unroll
  for (int p = 0; p < 2; ++p) {
    const int t0 = wave + p * 16;           // 0..7 or 16..23 : always valid
    const int t1 = wave + 8 + p * 16;       // 8..15 or 24..31
    const bool v1 = (t1 < TOTAL_TILES);
    const int t1c = v1 ? t1 : (TOTAL_TILES - 1);

    const float* arow0 = obs_b + (size_t)frame_of(t0)  * ND;
    const float* arow1 = obs_b + (size_t)frame_of(t1c) * ND;

    v8f acc0[4] = {vz, vz, vz, vz};
    v8f acc1[4] = {vz, vz, vz, vz};

    for (int kb = 0; kb < ND; kb += 4) {
      const int k0 = kb + 2 * half;
      const float2 ap0 = *(const float2*)(arow0 + k0);
      const float2 ap1 = *(const float2*)(arow1 + k0);
      v2f a0; a0.x = ap0.x; a0.y = ap0.y;
      v2f a1; a1.x = ap1.x; a1.y = ap1.y;
      const float* w0 = w_red + (size_t)k0 * NR;   // B[k0][*]
      const float* w1 = w0 + NR;                   // B[k0+1][*]
      v2f bf[4];
#pragma unroll
      for (int nt = 0; nt < 4; ++nt) {
        bf[nt].x = w0[nt * 16 + ln16];
        bf[nt].y = w1[nt * 16 + ln16];
      }
#pragma unroll
      for (int nt = 0; nt < 4; ++nt) {
        acc0[nt] = wmma_f32_4(a0, bf[nt], acc0[nt]);
        acc1[nt] = wmma_f32_4(a1, bf[nt], acc1[nt]);
      }
    }

    epilogue(acc0, t0);
    if (v1) epilogue(acc1, t1);
  }

  // combine halves, then cross-wave max via LDS
#pragma unroll
  for (int nt = 0; nt < 4; ++nt)
    lmax[nt] = fmaxf(lmax[nt], __shfl_xor(lmax[nt], 16, 32));
  if (lane < 16) {
#pragma unroll
    for (int nt = 0; nt < 4; ++nt)
      smax[wave * NR + nt * 16 + ln16] = lmax[nt];
  }
  __syncthreads();
  if (tid < NR) {
    float m = smax[tid];
#pragma unroll
    for (int w = 1; w < 8; ++w) m = fmaxf(m, smax[w * NR + tid]);
    ws_past[(size_t)b * NR + tid] = m;
  }
}

// ---------------------------------------------------------------------------
// Kernel 2: cumulative max over [past_pooled ; present] (21 -> keep last 20)
// ---------------------------------------------------------------------------
__global__ __launch_bounds__(64) void k_scan(
    const float* __restrict__ ws_past, const float* __restrict__ ws_present,
    float* __restrict__ ws_gm)        // [NB][NLOCAL][NR]
{
  const int b = blockIdx.x;
  const int r = threadIdx.x;
  float run = ws_past[(size_t)b * NR + r];
  for (int l = 0; l < NLOCAL; ++l) {
    const size_t idx = ((size_t)(b * NLOCAL + l)) * NR + r;
    run = fmaxf(run, ws_present[idx]);
    ws_gm[idx] = run;
  }
}

// ---------------------------------------------------------------------------
// Kernel 3: global_max = LN(relu(gm @ w_exp + b_exp)); rows=640, K=64, N=512.
// One block per 16-row M-tile; 32 waves, one 16-col N-tile each; LDS atomics
// accumulate the 512-wide row moments for the LayerNorm.
// ---------------------------------------------------------------------------
__global__ __launch_bounds__(1024) void k_expand(
    const float* __restrict__ ws_gm, const float* __restrict__ w_exp,
    const float* __restrict__ b_exp, const float* __restrict__ g_exp,
    const float* __restrict__ beta_exp,
    float* __restrict__ out)          // [NB*NLOCAL][ND]
{
  __shared__ float ssum[16];
  __shared__ float sssq[16];

  const int mt   = blockIdx.x;
  const int tid  = threadIdx.x;
  const int nt   = tid >> 5;          // N-tile (0..31)
  const int lane = tid & 31;
  const int half = lane >> 4;
  const int ln16 = lane & 15;
  const int dcol = nt * 16 + ln16;

  if (tid < 16) { ssum[tid] = 0.f; sssq[tid] = 0.f; }
  __syncthreads();

  const float* arow = ws_gm + (size_t)(mt * 16 + ln16) * NR;

  v8f acc = {0.f,0.f,0.f,0.f,0.f,0.f,0.f,0.f};
#pragma unroll
  for (int kb = 0; kb < NR; kb += 4) {
    const int k0 = kb + 2 * half;
    const float2 ap = *(const float2*)(arow + k0);
    v2f a; a.x = ap.x; a.y = ap.y;
    v2f bb;
    bb.x = w_exp[(size_t)k0 * ND + dcol];
    bb.y = w_exp[(size_t)(k0 + 1) * ND + dcol];
    acc = wmma_f32_4(a, bb, acc);
  }

  const float be = b_exp[dcol], ge = g_exp[dcol], bt = beta_exp[dcol];
  float x[8];
#pragma unroll
  for (int i = 0; i < 8; ++i) {
    x[i] = fmaxf(acc[i] + be, 0.f);
    float s = x[i], ss = x[i] * x[i];
#pragma unroll
    for (int d = 1; d < 16; d <<= 1) {
      s  += __shfl_xor(s,  d, 32);
      ss += __shfl_xor(ss, d, 32);
    }
    if (ln16 == 0) {
      const int row = i + 8 * half;
      atomicAdd(&ssum[row], s);
      atomicAdd(&sssq[row], ss);
    }
  }
  __syncthreads();

#pragma unroll
  for (int i = 0; i < 8; ++i) {
    const int row  = i + 8 * half;
    const float mean = ssum[row] * (1.f / 512.f);
    const float var  = sssq[row] * (1.f / 512.f) - mean * mean;
    const float rinv = rsqrtf(var + LN_EPS);
    const float y = (x[i] - mean) * rinv * ge + bt;
    const int p = mt * 16 + row;
    out[(size_t)p * ND + dcol] = y;
  }
}

// ---------------------------------------------------------------------------
extern "C" void kernel_launch(void* const* d_in, const int* in_sizes, int n_in,
                              void* d_out, int out_size, void* d_ws,
                              size_t ws_size, hipStream_t stream) {
  (void)in_sizes; (void)n_in; (void)out_size; (void)ws_size;

  const float* obs      = (const float*)d_in[0];
  const float* w_red    = (const float*)d_in[1];
  const float* b_red    = (const float*)d_in[2];
  const float* g_red    = (const float*)d_in[3];
  const float* beta_red = (const float*)d_in[4];
  const float* w_exp    = (const float*)d_in[5];
  const float* b_exp    = (const float*)d_in[6];
  const float* g_exp    = (const float*)d_in[7];
  const float* beta_exp = (const float*)d_in[8];

  float* out     = (float*)d_out;
  float* out_now = out + (size_t)NB * NLOCAL * ND;   // second tuple output

  float* ws         = (float*)d_ws;
  float* ws_past    = ws;                                // NB*NR
  float* ws_present = ws_past + (size_t)NB * NR;         // NB*NLOCAL*NR
  float* ws_gm      = ws_present + (size_t)NB * NLOCAL * NR;

  k_reduce<<<NB, 256, 0, stream>>>(obs, w_red, b_red, g_red, beta_red,
                                   ws_past, ws_present, out_now);
  k_scan<<<NB, 64, 0, stream>>>(ws_past, ws_present, ws_gm);
  k_expand<<<(NB * NLOCAL) / 16, 1024, 0, stream>>>(ws_gm, w_exp, b_exp,
                                                    g_exp, beta_exp, out);
}